// AutoEncoder_68719477503
// MI455X (gfx1250) — compile-verified
//
#include <hip/hip_runtime.h>

typedef __attribute__((ext_vector_type(2))) float v2f;
typedef __attribute__((ext_vector_type(8))) float v8f;

#define NQ      8      // data qubits
#define DIM     256    // 2^8
#define DEPTH   4
#define MROWS   128    // stacked [Re;Im] of the 64 needed rows
#define BATCH   4096   // fixed by the reference

// ---------------------------------------------------------------------------
// Kernel 1: build A (128 x 256, row-major).
// Block jj simulates the entangler circuit on basis state e_jj (256 complex
// amplitudes in LDS), then writes column jj:
//   A[x     ][jj] = Re( U[4x][jj] * (-i)^pop(jj) )
//   A[64 + x][jj] = Im( U[4x][jj] * (-i)^pop(jj) )
// Wire w <-> bit (7-w) of the flat index (wire 0 = MSB).
// ---------------------------------------------------------------------------
__global__ __launch_bounds__(64) void build_A(const float* __restrict__ weights,
                                              float* __restrict__ A) {
    __shared__ float sr[DIM];
    __shared__ float si[DIM];
    const int jj = blockIdx.x;      // column / basis state, 0..255
    const int t  = threadIdx.x;     // 0..63

    for (int i = t; i < DIM; i += 64) { sr[i] = (i == jj) ? 1.0f : 0.0f; si[i] = 0.0f; }
    __syncthreads();

    for (int l = 0; l < DEPTH; ++l) {
        // --- RX(W[l,w]) on wire w, w = 0..7 (commuting singles) ---
        for (int w = 0; w < NQ; ++w) {
            const float th = weights[l * NQ + w] * 0.5f;
            const float c = cosf(th), s = sinf(th);
            const int mask = 1 << (7 - w);
            #pragma unroll
            for (int r = 0; r < 2; ++r) {        // 128 pairs / 64 threads
                const int p  = t + r * 64;
                const int lo = p & (mask - 1);
                const int hi = (p ^ lo) << 1;
                const int i0 = hi | lo;
                const int i1 = i0 | mask;
                const float a0r = sr[i0], a0i = si[i0];
                const float a1r = sr[i1], a1i = si[i1];
                // RX = [[c, -is],[-is, c]]
                sr[i0] =  c * a0r + s * a1i;
                si[i0] =  c * a0i - s * a1r;
                sr[i1] =  s * a0i + c * a1r;
                si[i1] = -s * a0r + c * a1i;
            }
            __syncthreads();
        }
        // --- CNOT ring: ctrl w, target (w+1)%8, sequential in w ---
        for (int w = 0; w < NQ; ++w) {
            const int pc = 7 - w;
            const int pt = 7 - ((w + 1) & 7);
            const int mc   = 1 << pc;
            const int mt_  = 1 << pt;
            const int mmin = (mc < mt_) ? mc  : mt_;
            const int mmax = (mc < mt_) ? mt_ : mc;
            // spread t's 6 bits into the 6 free positions
            int x = t;
            x = ((x & ~(mmin - 1)) << 1) | (x & (mmin - 1));
            x = ((x & ~(mmax - 1)) << 1) | (x & (mmax - 1));
            const int i0 = x | mc;          // ctrl=1, tgt=0
            const int i1 = i0 | mt_;        // ctrl=1, tgt=1
            const float r0 = sr[i0], q0 = si[i0];
            sr[i0] = sr[i1]; si[i0] = si[i1];
            sr[i1] = r0;     si[i1] = q0;
            __syncthreads();
        }
    }

    // phase (-i)^pop(jj), keep rows 4x, write column jj of A
    const int ph = __popc(jj) & 3;
    const float zr = sr[4 * t], zi = si[4 * t];
    float vr, vi;
    switch (ph) {
        case 0:  vr =  zr; vi =  zi; break;
        case 1:  vr =  zi; vi = -zr; break;   // z * (-i)
        case 2:  vr = -zr; vi = -zi; break;   // z * (-1)
        default: vr = -zi; vi =  zr; break;   // z * (+i)
    }
    A[t * DIM + jj]        = vr;
    A[(64 + t) * DIM + jj] = vi;
}

// ---------------------------------------------------------------------------
// Kernel 2: build Phi, PACKED fragment-major for the wmma B operand:
//   Phi_p[ (p*BATCH + b)*2 + {0,1} ] = { m(2p, b), m(2p+1, b) },  p = 0..127
// where m(j,b) = prod over bits k of j: bit ? sin(f[b,7-k]/2) : cos(f[b,7-k]/2)
// Each thread emits coalesced 8-byte stores (consecutive b -> consecutive).
// ---------------------------------------------------------------------------
__global__ __launch_bounds__(256) void build_phi(const float* __restrict__ feat,
                                                 float* __restrict__ Phi_p) {
    const int b = blockIdx.x * blockDim.x + threadIdx.x;
    if (b >= BATCH) return;
    float cw[NQ], sw[NQ];
    #pragma unroll
    for (int w = 0; w < NQ; ++w) {
        const float th = feat[b * NQ + w] * 0.5f;
        cw[w] = cosf(th);
        sw[w] = sinf(th);
    }
    for (int j = 0; j < DIM; j += 2) {
        float mbase = 1.0f;                   // product over bits 1..7 of j
        #pragma unroll
        for (int k = 1; k < NQ; ++k)
            mbase *= ((j >> k) & 1) ? sw[7 - k] : cw[7 - k];
        v2f pair;
        pair.x = mbase * cw[7];               // bit0 = 0  (wire 7)
        pair.y = mbase * sw[7];               // bit0 = 1
        *(v2f*)(Phi_p + ((size_t)(j >> 1) * BATCH + b) * 2) = pair;
    }
}

// ---------------------------------------------------------------------------
// Kernel 3: Z = A * Phi via V_WMMA_F32_16X16X4_F32, fused column reduction:
//   out[b] = 0.5 * (1 - sum_i Z[i,b]^2)
// One wave per 16-column tile. kk outer / mt inner with 8 accumulators:
// per K-step = 1 B-frag b64 load + 8 A-frag b64 loads + 8 wmma, all loads
// from two fixed base addresses with compile-time immediate offsets.
// ---------------------------------------------------------------------------
__global__ __launch_bounds__(256) void gemm_reduce(const float* __restrict__ A,
                                                   const float* __restrict__ Phi_p,
                                                   float* __restrict__ out) {
    const int lane = threadIdx.x & 31;
    const int wave = threadIdx.x >> 5;
    const int nt   = blockIdx.x * 8 + wave;      // N-tile (16 columns)
    const int half = lane >> 4;
    const int l16  = lane & 15;
    const int n    = nt * 16 + l16;

    // per-lane fixed bases; everything else is an immediate offset
    const float* abase = A + l16 * DIM + half * 2;                    // + mt*16*DIM + kk*4
    const float* bbase = Phi_p + ((size_t)half * BATCH + n) * 2;      // + kk*2*BATCH*2

    v8f acc[8];
    #pragma unroll
    for (int mt = 0; mt < 8; ++mt) acc[mt] = (v8f){};

    #pragma unroll 2
    for (int kk = 0; kk < 64; ++kk) {
        const v2f bf = *(const v2f*)(bbase + (size_t)kk * (2 * BATCH * 2));
        #pragma unroll
        for (int mt = 0; mt < 8; ++mt) {
            const v2f a = *(const v2f*)(abase + mt * 16 * DIM + kk * 4);
            acc[mt] = __builtin_amdgcn_wmma_f32_16x16x4_f32(
                          /*neg_a=*/false, a, /*neg_b=*/false, bf,
                          /*c_mod=*/(short)0, acc[mt],
                          /*reuse_a=*/false, /*reuse_b=*/false);
        }
    }

    float ssum = 0.0f;
    #pragma unroll
    for (int mt = 0; mt < 8; ++mt)
        #pragma unroll
        for (int r = 0; r < 8; ++r) ssum += acc[mt][r] * acc[mt][r];

    // lane L and L^16 hold the two M-halves for the same column n
    const float tot = ssum + __shfl_xor(ssum, 16, 32);
    if (lane < 16) out[n] = 0.5f * (1.0f - tot);
}

// ---------------------------------------------------------------------------
extern "C" void kernel_launch(void* const* d_in, const int* in_sizes, int n_in,
                              void* d_out, int out_size, void* d_ws, size_t ws_size,
                              hipStream_t stream) {
    const float* features = (const float*)d_in[0];   // (4096, 8) f32
    const float* weights  = (const float*)d_in[1];   // (32,)    f32
    float* out = (float*)d_out;                      // (4096,)  f32

    float* A     = (float*)d_ws;                                           // 128*256 f32
    float* Phi_p = (float*)((char*)d_ws + (size_t)MROWS * DIM * sizeof(float)); // 256*4096 f32

    build_A<<<DIM, 64, 0, stream>>>(weights, A);
    build_phi<<<BATCH / 256, 256, 0, stream>>>(features, Phi_p);
    gemm_reduce<<<BATCH / 128, 256, 0, stream>>>(A, Phi_p, out);
}